// e3jLayer_55147380081187
// MI455X (gfx1250) — compile-verified
//
#include <hip/hip_runtime.h>
#include <math.h>

// ---------------------------------------------------------------------------
// E(3)-equivariant layer for MI455X (gfx1250, wave32).
//  K1 build_w:  Racah CG -> real-basis TP tensor W[9][9][9] into d_ws (2.9 KB)
//  K2 zero:     zero d_out (used as the segment-sum accumulator)
//  K3 edge:     sh(L=2) -> T=sh*W -> msg = x_send @ T -> atomicAdd to out
//  K4 dense:    per node 16x9 @ 9x9 + bias, in place, via 3x V_WMMA_F32_16X16X4_F32
// ---------------------------------------------------------------------------

typedef __attribute__((ext_vector_type(2))) float v2f;
typedef __attribute__((ext_vector_type(8))) float v8f;

#define IRR 9          // (L+1)^2, L=2
#define NCH 16         // channels
#define TILE (NCH*IRR) // 144 floats per node

// ------------------------------ K1: build W --------------------------------

__device__ __constant__ double c_fact[9] = {1.,1.,2.,6.,24.,120.,720.,5040.,40320.};

// complex Clebsch-Gordan <j1 m1 j2 m2 | j3 m3> (Racah), real-valued
__device__ double cg_complex(int j1, int j2, int j3, int m1, int m2, int m3) {
  if (m1 + m2 != m3) return 0.0;
  double pref = sqrt((2.0*j3+1.0) * c_fact[j1+j2-j3] * c_fact[j1-j2+j3] *
                     c_fact[-j1+j2+j3] / c_fact[j1+j2+j3+1]);
  pref *= sqrt(c_fact[j1+m1]*c_fact[j1-m1]*c_fact[j2+m2]*c_fact[j2-m2]*
               c_fact[j3+m3]*c_fact[j3-m3]);
  double s = 0.0;
  for (int k = 0; k <= j1+j2-j3; ++k) {
    int d2 = j1+j2-j3-k, d3 = j1-m1-k, d4 = j2+m2-k, d5 = j3-j2+m1+k, d6 = j3-j1-m2+k;
    if (d2 < 0 || d3 < 0 || d4 < 0 || d5 < 0 || d6 < 0) continue;
    double denom = c_fact[k]*c_fact[d2]*c_fact[d3]*c_fact[d4]*c_fact[d5]*c_fact[d6];
    s += ((k & 1) ? -1.0 : 1.0) / denom;
  }
  return pref * s;
}

// complex->real SH basis matrix entry A[a][mi] (mi = m + l), as in _real_mat
__device__ void amat(int l, int a, int mi, double* re, double* im) {
  const double s = 0.70710678118654752440;
  *re = 0.0; *im = 0.0;
  int m  = mi - l;   // complex m
  int ar = a - l;    // signed real row
  if (ar == 0) { if (m == 0) *re = 1.0; return; }
  if (ar > 0) {                       // cosine-type row (l+m)
    if (m ==  ar) *re = (ar & 1) ? -s : s;   // (-1)^m * s
    else if (m == -ar) *re = s;
  } else {                            // sine-type row (l-m)
    int mm = -ar;
    if (m == -mm) *im = s;                    //  i*s
    else if (m == mm) *im = (mm & 1) ? s : -s; // -i*(-1)^m*s
  }
}

__global__ void build_w_kernel(float* __restrict__ Wg) {
  const int tid = threadIdx.x;
  // zero the full 9x9x9 tensor first (paths write disjoint blocks over it)
  for (int i = tid; i < IRR*IRR*IRR; i += blockDim.x) Wg[i] = 0.f;
  __syncthreads();

  // all (l1,l2)->l3<=2 paths, |l1-l2|<=l3<=l1+l2
  const int P[15][3] = {{0,0,0},{0,1,1},{0,2,2},{1,0,1},{1,1,0},{1,1,1},{1,1,2},
                        {1,2,1},{1,2,2},{2,0,2},{2,1,1},{2,1,2},{2,2,0},{2,2,1},{2,2,2}};
  const int NWORK = 615; // sum of block sizes (2l1+1)(2l2+1)(2l3+1)

  for (int t = tid; t < NWORK; t += blockDim.x) {
    int rem = t, l1 = 0, l2 = 0, l3 = 0, sz = 0;
    for (int p = 0; p < 15; ++p) {
      l1 = P[p][0]; l2 = P[p][1]; l3 = P[p][2];
      sz = (2*l1+1)*(2*l2+1)*(2*l3+1);
      if (rem < sz) break;
      rem -= sz;
    }
    const int d3 = 2*l3+1, d2 = 2*l2+1;
    const int cidx = rem % d3;
    const int bidx = (rem / d3) % d2;
    const int aidx = rem / (d3*d2);

    double accR = 0.0, accI = 0.0;
    for (int m = -l1; m <= l1; ++m) {
      for (int n = -l2; n <= l2; ++n) {
        const int o = m + n;
        if (o < -l3 || o > l3) continue;
        const double w = cg_complex(l1, l2, l3, m, n, o);
        if (w == 0.0) continue;
        double r1, i1, r2, i2, r3, i3;
        amat(l1, aidx, m + l1, &r1, &i1); i1 = -i1;   // conj
        amat(l2, bidx, n + l2, &r2, &i2); i2 = -i2;   // conj
        amat(l3, cidx, o + l3, &r3, &i3);
        const double pr = r1*r2 - i1*i2;
        const double pi = r1*i2 + i1*r2;
        accR += w * (pr*r3 - pi*i3);
        accI += w * (pr*i3 + pi*r3);
      }
    }
    // parity selects which part is (identically) nonzero == reference's max-abs rule
    const double val = ((l1 + l2 + l3) & 1) ? accI : accR;
    Wg[(l1*l1 + aidx)*81 + (l2*l2 + bidx)*9 + (l3*l3 + cidx)] = (float)val;
  }
}

// ------------------------------ K2: zero -----------------------------------

__global__ void zero_kernel(float* __restrict__ p, size_t n) {
  size_t i = (size_t)blockIdx.x * blockDim.x + threadIdx.x;
  const size_t stride = (size_t)gridDim.x * blockDim.x;
  for (; i < n; i += stride) p[i] = 0.f;
}

// ------------------------------ K3: edges ----------------------------------

__launch_bounds__(256)
__global__ void edge_kernel(const float* __restrict__ pos,
                            const float* __restrict__ feats,
                            const int*  __restrict__ snd,
                            const int*  __restrict__ rcv,
                            const float* __restrict__ Wg,
                            float* __restrict__ agg, int nE) {
  __shared__ float Ws[IRR*IRR*IRR];
  for (int i = threadIdx.x; i < IRR*IRR*IRR; i += blockDim.x) Ws[i] = Wg[i];
  __syncthreads();

  const int e = blockIdx.x * blockDim.x + threadIdx.x;
  if (e >= nE) return;

  const int s = snd[e], r = rcv[e];
  const float dx = pos[3*r+0] - pos[3*s+0];
  const float dy = pos[3*r+1] - pos[3*s+1];
  const float dz = pos[3*r+2] - pos[3*s+2];
  const float inv = 1.f / sqrtf(dx*dx + dy*dy + dz*dz + 1e-12f);

  float sh[IRR];
  sh[0] = 1.f;
  sh[1] = dy * inv;   // y1 = (ry, rz, rx)
  sh[2] = dz * inv;
  sh[3] = dx * inv;

  // l=2 block from CG(1,1,2) == W[1:4,1:4,4:9], then renormalize (matches ref)
  float q[5];
  #pragma unroll
  for (int c = 0; c < 5; ++c) {
    float acc = 0.f;
    #pragma unroll
    for (int a = 0; a < 3; ++a)
      #pragma unroll
      for (int b = 0; b < 3; ++b)
        acc = fmaf(sh[1+a] * sh[1+b], Ws[(1+a)*81 + (1+b)*9 + (4+c)], acc);
    q[c] = acc;
  }
  float qn = q[0]*q[0] + q[1]*q[1] + q[2]*q[2] + q[3]*q[3] + q[4]*q[4];
  qn = 1.f / sqrtf(qn + 1e-12f);
  #pragma unroll
  for (int c = 0; c < 5; ++c) sh[4+c] = q[c] * qn;

  // T[i][k] = sum_j sh[j] * W[i][j][k]   (LDS broadcast reads, 729 FMAs)
  float T[IRR*IRR];
  #pragma unroll
  for (int i = 0; i < IRR; ++i)
    #pragma unroll
    for (int k = 0; k < IRR; ++k) {
      float acc = 0.f;
      #pragma unroll
      for (int j = 0; j < IRR; ++j)
        acc = fmaf(sh[j], Ws[(i*IRR + j)*IRR + k], acc);
      T[i*IRR + k] = acc;
    }

  const float* __restrict__ xb = feats + (size_t)s * TILE;
  float* __restrict__ ab = agg + (size_t)r * TILE;
  for (int c = 0; c < NCH; ++c) {          // keep channel loop rolled (I$)
    float x[IRR];
    #pragma unroll
    for (int i = 0; i < IRR; ++i) x[i] = xb[c*IRR + i];
    #pragma unroll
    for (int k = 0; k < IRR; ++k) {
      float m = 0.f;
      #pragma unroll
      for (int i = 0; i < IRR; ++i) m = fmaf(x[i], T[i*IRR + k], m);
      unsafeAtomicAdd(&ab[c*IRR + k], m);  // global_atomic_add_f32
    }
  }
}

// ------------------------------ K4: dense (WMMA) ---------------------------
// Per node: D(16x16) = A(16x12) x B(12x16) + bias, via 3x V_WMMA_F32_16X16X4_F32.
// A rows = 16 channels, K = 9 irreps (zero-padded to 12), N = 9 (padded to 16).
// In-place on d_out: each wave reads its whole 16x9 tile before storing.

__launch_bounds__(256)
__global__ void dense_kernel(const float* __restrict__ dw,
                             const float* __restrict__ db,
                             float* __restrict__ out, int nNodes) {
  __shared__ float Bw[12*16];   // dense_w zero-padded to 12(K) x 16(N)
  __shared__ float bias[16];
  const int tid = threadIdx.x;
  if (tid < 192) {
    const int k = tid / 16, n = tid % 16;
    Bw[tid] = (k < IRR && n < IRR) ? dw[k*IRR + n] : 0.f;
  }
  if (tid < 16) bias[tid] = (tid < IRR) ? db[tid] : 0.f;
  __syncthreads();

  const int wave = tid >> 5;
  const int lane = tid & 31;
  const int node = blockIdx.x * 8 + wave;
  if (node >= nNodes) return;            // whole-wave exit: EXEC stays all-1s

  float* __restrict__ tile = out + (size_t)node * TILE;
  const int half = lane >> 4;            // 0: K pair (k0,k0+1), 1: (k0+2,k0+3)
  const int l16  = lane & 15;            // A: row M ; B/D: column N

  v2f a[3], bf[3];
  #pragma unroll
  for (int kk = 0; kk < 3; ++kk) {
    const int k0 = kk*4 + half*2;
    a[kk].x = (k0     < IRR) ? tile[l16*IRR + k0    ] : 0.f;
    a[kk].y = (k0 + 1 < IRR) ? tile[l16*IRR + k0 + 1] : 0.f;
    bf[kk].x = Bw[(k0    )*16 + l16];
    bf[kk].y = Bw[(k0 + 1)*16 + l16];
  }

  v8f c;
  #pragma unroll
  for (int r = 0; r < 8; ++r) c[r] = bias[l16];

  c = __builtin_amdgcn_wmma_f32_16x16x4_f32(false, a[0], false, bf[0], (short)0, c, false, false);
  c = __builtin_amdgcn_wmma_f32_16x16x4_f32(false, a[1], false, bf[1], (short)0, c, false, false);
  c = __builtin_amdgcn_wmma_f32_16x16x4_f32(false, a[2], false, bf[2], (short)0, c, false, false);

  if (l16 < IRR) {
    #pragma unroll
    for (int r = 0; r < 8; ++r) {
      const int M = r + 8*half;          // C/D layout: VGPR r -> M=r / M=r+8
      tile[M*IRR + l16] = c[r];
    }
  }
}

// ------------------------------ launch -------------------------------------

extern "C" void kernel_launch(void* const* d_in, const int* in_sizes, int n_in,
                              void* d_out, int out_size, void* d_ws, size_t ws_size,
                              hipStream_t stream) {
  const float* pos   = (const float*)d_in[0];
  const float* feats = (const float*)d_in[1];
  const float* dw    = (const float*)d_in[2];
  const float* db    = (const float*)d_in[3];
  const int*   snd   = (const int*)d_in[4];
  const int*   rcv   = (const int*)d_in[5];
  float* out = (float*)d_out;
  float* Wg  = (float*)d_ws;             // 729 floats of workspace

  const int nE = in_sizes[4];
  const int nN = in_sizes[0] / 3;

  build_w_kernel<<<1, 256, 0, stream>>>(Wg);
  zero_kernel<<<2048, 256, 0, stream>>>(out, (size_t)out_size);
  edge_kernel<<<(nE + 255) / 256, 256, 0, stream>>>(pos, feats, snd, rcv, Wg, out, nE);
  dense_kernel<<<(nN + 7) / 8, 256, 0, stream>>>(dw, db, out, nN);
}